// WindowingLayer_22084721836441
// MI455X (gfx1250) — compile-verified
//
#include <hip/hip_runtime.h>

// ---------------------------------------------------------------------------
// Windowing layer: out[b,i,w,f] = x[b,i+w,f]  (zero-padded past S)
//   x:   [32, 2048, 128]  f32
//   out: [32, 2048, 16, 128] f32   (512 MiB -> write-bandwidth bound, ~23us)
// Strategy: stage input rows once in LDS (async DMA global->LDS), fan out 16
// window copies per row with async LDS->global stores. 512B coalesced per
// wave-instruction, ASYNCcnt-tracked (CDNA5 async data-movement path).
// ---------------------------------------------------------------------------

#define HAS_ASYNC_LOAD  __has_builtin(__builtin_amdgcn_global_load_async_to_lds_b128)
#define HAS_ASYNC_STORE __has_builtin(__builtin_amdgcn_global_store_async_from_lds_b128)

typedef int v4i __attribute__((ext_vector_type(4)));
typedef __attribute__((address_space(1))) v4i gv4i;   // global (AS1)
typedef __attribute__((address_space(3))) v4i lv4i;   // LDS    (AS3)

constexpr int B  = 32;
constexpr int S  = 2048;
constexpr int F  = 128;
constexpr int W  = 16;
constexpr int TB = 32;             // output positions per block
constexpr int ROWS = TB + W - 1;   // 47 input rows staged in LDS
constexpr int NT = 256;            // 8 wave32
constexpr int F4 = F / 4;          // 32 float4 chunks per row (512B row)

__device__ __forceinline__ void wait_async0() {
#if __has_builtin(__builtin_amdgcn_s_wait_asynccnt)
    __builtin_amdgcn_s_wait_asynccnt(0);
#else
    asm volatile("s_wait_asynccnt 0" ::: "memory");
#endif
}

__global__ __launch_bounds__(NT) void windowing_kernel(
        const float* __restrict__ x, float* __restrict__ out) {
    __shared__ float smem[ROWS * F];   // 24,064 B

    const int b   = blockIdx.x / (S / TB);
    const int i0  = (blockIdx.x % (S / TB)) * TB;
    const int tid = threadIdx.x;

    // ---- Phase 1: stage x[b, i0 : i0+ROWS, :] into LDS (zeros past S) ----
    for (int c = tid; c < ROWS * F4; c += NT) {
        const int row = c >> 5;            // c / F4
        const int l4  = c & (F4 - 1);
        const int ig  = i0 + row;
        float* lp = &smem[row * F + l4 * 4];
        if (ig < S) {
            const float* gp = x + ((size_t)b * S + ig) * F + l4 * 4;
#if HAS_ASYNC_LOAD
            __builtin_amdgcn_global_load_async_to_lds_b128(
                (gv4i*)gp, (lv4i*)lp, 0, 0);
#else
            *(float4*)lp = *(const float4*)gp;
#endif
        } else {
            *(float4*)lp = make_float4(0.f, 0.f, 0.f, 0.f);
        }
    }
    wait_async0();        // our wave's async LDS writes have landed
    __syncthreads();      // all waves' staging complete

    // ---- Phase 2: emit out[b, i0+r, w, :] = LDS row (r+w) ----
    // One wave handles one 512B output row per issued instruction:
    // lane k covers bytes [16k, 16k+16) of the row -> fully coalesced.
    const int wave = tid >> 5;
    const int lane = tid & 31;
    for (int orow = wave; orow < TB * W; orow += (NT / 32)) {
        const int r = orow >> 4;           // orow / W
        const int w = orow & (W - 1);
        const float* lp = &smem[(r + w) * F + lane * 4];
        float* gp = out + (((size_t)b * S + (i0 + r)) * W + w) * F + lane * 4;
#if HAS_ASYNC_STORE
        __builtin_amdgcn_global_store_async_from_lds_b128(
            (gv4i*)gp, (lv4i*)lp, 0, 0);
#else
        *(float4*)gp = *(const float4*)lp;
#endif
    }
    wait_async0();        // drain async stores before wave retires
}

extern "C" void kernel_launch(void* const* d_in, const int* in_sizes, int n_in,
                              void* d_out, int out_size, void* d_ws, size_t ws_size,
                              hipStream_t stream) {
    const float* x  = (const float*)d_in[0];
    float* out      = (float*)d_out;
    dim3 grid(B * (S / TB));   // 2048 blocks
    windowing_kernel<<<grid, NT, 0, stream>>>(x, out);
}